// SpKBGATModified_53807350284809
// MI455X (gfx1250) — compile-verified
//
#include <hip/hip_runtime.h>
#include <hip/hip_bf16.h>

// ---------------------------------------------------------------------------
// KBGAT forward for MI455X (gfx1250, wave32).
// Dense GEMMs use V_WMMA_F32_16X16X4_F32 (fp32 in, fp32 acc) — the problem is
// memory/atomic bound (~0.5GB gather + ~150M f32 atomics vs ~70 GFLOP dense),
// so fp32 WMMA keeps full precision at zero cost.
// GEMM inner loop uses clamped (not predicated) addressing so loads lower to
// plain global_load_b32 with no EXEC save/restore or per-load waits.
// ---------------------------------------------------------------------------

typedef __attribute__((ext_vector_type(2))) float v2f;
typedef __attribute__((ext_vector_type(8))) float v8f;

#define NN      50000   // N_NODES
#define NR      500     // N_REL
#define NE_MAIN 200000  // E
#define NE_NEW  100000  // E_NEW
#define NE_NHOP 50000   // E_NHOP
#define NBATCH  4096
#define DI      50      // D_IN
#define DH      100     // D_HID
#define DO      200     // D_OUT
#define ALPHA_LRELU 0.2f

static inline int ceil_div(int a, int b) { return (a + b - 1) / b; }

// ---------------------------------------------------------------------------
// fp32 WMMA GEMM: C[M,N] = A[M,K] * B
//   TRANSB==0 : B is [K,N] row-major, row stride ldb   (B(k,n)=B[k*ldb+n])
//   TRANSB==1 : B is [N,K] row-major, row stride ldb   (B(k,n)=B[n*ldb+k])
// One wave computes a 16x32 strip (two 16x16 tiles sharing the A fragment).
// Out-of-range rows/cols use clamped addresses (safe loads, masked stores);
// only the K-tail needs value masking, and only on the A operand.
// ---------------------------------------------------------------------------
template <int TRANSB>
__global__ __launch_bounds__(256) void wmma_gemm_f32(
    const float* __restrict__ A, const float* __restrict__ B, float* __restrict__ C,
    int M, int N, int K, int lda, int ldb, int ldc) {
  const int wave = threadIdx.x >> 5;
  const int lane = threadIdx.x & 31;
  const int tileM = (blockIdx.x * 8 + wave) * 16;
  const int tileN = blockIdx.y * 32;
  if (tileM >= M || tileN >= N) return;  // wave-uniform exit

  // A fragment (16x4): lanes 0-15 -> M=lane, K={0,1}; lanes 16-31 -> K={2,3}
  // B fragment (4x16): lanes 0-15 -> N=lane, K={0,1}; lanes 16-31 -> K={2,3}
  const int mA  = lane & 15;
  const int kHi = (lane >> 4) * 2;  // 0 or 2
  const int nB  = lane & 15;

  const int gm  = tileM + mA;
  const int gn0 = tileN + nB;
  const int gn1 = tileN + 16 + nB;
  const int gmc  = (gm  < M) ? gm  : (M - 1);
  const int gn0c = (gn0 < N) ? gn0 : (N - 1);
  const int gn1c = (gn1 < N) ? gn1 : (N - 1);

  const float* Ap = A + (long)gmc * lda + kHi;
  const float* Bp0;
  const float* Bp1;
  if (TRANSB) {
    Bp0 = B + (long)gn0c * ldb + kHi;
    Bp1 = B + (long)gn1c * ldb + kHi;
  } else {
    Bp0 = B + (long)kHi * ldb + gn0c;
    Bp1 = B + (long)kHi * ldb + gn1c;
  }

  v8f c0 = {0.f, 0.f, 0.f, 0.f, 0.f, 0.f, 0.f, 0.f};
  v8f c1 = {0.f, 0.f, 0.f, 0.f, 0.f, 0.f, 0.f, 0.f};

  const int K4 = K & ~3;
  for (int k0 = 0; k0 < K4; k0 += 4) {
    v2f a, b0, b1;
    a.x = Ap[k0];
    a.y = Ap[k0 + 1];
    if (TRANSB) {
      b0.x = Bp0[k0];  b0.y = Bp0[k0 + 1];
      b1.x = Bp1[k0];  b1.y = Bp1[k0 + 1];
    } else {
      b0.x = Bp0[(long)k0 * ldb];        b0.y = Bp0[(long)(k0 + 1) * ldb];
      b1.x = Bp1[(long)k0 * ldb];        b1.y = Bp1[(long)(k0 + 1) * ldb];
    }
    c0 = __builtin_amdgcn_wmma_f32_16x16x4_f32(false, a, false, b0, (short)0, c0, false, false);
    c1 = __builtin_amdgcn_wmma_f32_16x16x4_f32(false, a, false, b1, (short)0, c1, false, false);
  }
  if (K4 < K) {  // K-tail (e.g. K=50): clamp addresses, zero-mask A only
    const int ka0 = K4 + kHi, ka1 = ka0 + 1;
    const int kt0 = ((ka0 < K) ? ka0 : (K - 1)) - kHi;
    const int kt1 = ((ka1 < K) ? ka1 : (K - 1)) - kHi;
    const float av0 = Ap[kt0];
    const float av1 = Ap[kt1];
    v2f a, b0, b1;
    a.x = (ka0 < K) ? av0 : 0.f;
    a.y = (ka1 < K) ? av1 : 0.f;
    if (TRANSB) {
      b0.x = Bp0[kt0];  b0.y = Bp0[kt1];
      b1.x = Bp1[kt0];  b1.y = Bp1[kt1];
    } else {
      b0.x = Bp0[(long)kt0 * ldb];  b0.y = Bp0[(long)kt1 * ldb];
      b1.x = Bp1[(long)kt0 * ldb];  b1.y = Bp1[(long)kt1 * ldb];
    }
    c0 = __builtin_amdgcn_wmma_f32_16x16x4_f32(false, a, false, b0, (short)0, c0, false, false);
    c1 = __builtin_amdgcn_wmma_f32_16x16x4_f32(false, a, false, b1, (short)0, c1, false, false);
  }

  // C/D layout: VGPR v -> lanes 0-15: M=v, lanes 16-31: M=v+8; N = lane&15
  const int rowBase = tileM + (lane >> 4) * 8;
#pragma unroll
  for (int v = 0; v < 8; ++v) {
    const int row = rowBase + v;
    if (row < M) {
      if (gn0 < N) C[(long)row * ldc + gn0] = c0[v];
      if (gn1 < N) C[(long)row * ldc + gn1] = c1[v];
    }
  }
}

static void launch_gemm(bool transB, const float* A, const float* B, float* C,
                        int M, int N, int K, int lda, int ldb, int ldc,
                        hipStream_t s) {
  dim3 grid(ceil_div(M, 128), ceil_div(N, 32));
  if (transB)
    wmma_gemm_f32<1><<<grid, 256, 0, s>>>(A, B, C, M, N, K, lda, ldb, ldc);
  else
    wmma_gemm_f32<0><<<grid, 256, 0, s>>>(A, B, C, M, N, K, lda, ldb, ldc);
}

// ---------------------------------------------------------------------------
// Row l2 normalize (wave per row)
// ---------------------------------------------------------------------------
__global__ void l2norm_rows(const float* __restrict__ in, float* __restrict__ out,
                            int N, int D) {
  const int wave = threadIdx.x >> 5, lane = threadIdx.x & 31;
  const long row = (long)blockIdx.x * (blockDim.x >> 5) + wave;
  if (row >= N) return;
  const float* r = in + row * D;
  float ss = 0.f;
  for (int d = lane; d < D; d += 32) { float v = r[d]; ss += v * v; }
  for (int off = 16; off; off >>= 1) ss += __shfl_xor(ss, off);
  const float inv = 1.f / fmaxf(sqrtf(ss), 1e-12f);
  for (int d = lane; d < D; d += 32) out[row * D + d] = r[d] * inv;
}

// ---------------------------------------------------------------------------
// Sparse GAT edge kernels: one wave per edge (wave32).
// m = xs[src] + xd[dst] + er ;  s = m.a2 (+bias) ;  w = exp(-leakyrelu(s))
// scatter: denom[src] += w ;  H[src, slot, :] += w*m  (f32 atomics)
// ---------------------------------------------------------------------------
template <int D>
__global__ __launch_bounds__(256) void gat_edge_main(
    const float* __restrict__ xs, const float* __restrict__ xd,
    const float* __restrict__ relEr,  // [NR, D] precomputed eemb projection
    const int* __restrict__ srcA, const int* __restrict__ dstA,
    const int* __restrict__ etype, int E0,
    const int* __restrict__ nhop, int Enh,  // rows: [dst, et0, et1, src]
    const float* __restrict__ a2, float* __restrict__ H,
    float* __restrict__ denom, int slot, int nslots) {
  const int wave = threadIdx.x >> 5, lane = threadIdx.x & 31;
  const long e = (long)blockIdx.x * (blockDim.x >> 5) + wave;
  if (e >= (long)E0 + Enh) return;
  constexpr int NS = (D + 31) / 32;
  float m[NS];
  int s, d;
  if (e < E0) {
    s = srcA[e]; d = dstA[e];
    const float* er = relEr + (long)etype[e] * D;
#pragma unroll
    for (int i = 0; i < NS; ++i) { int dd = lane + i * 32; m[i] = (dd < D) ? er[dd] : 0.f; }
  } else {
    const long j = e - E0;
    s = nhop[j * 4 + 3]; d = nhop[j * 4 + 0];
    const float* er0 = relEr + (long)nhop[j * 4 + 1] * D;
    const float* er1 = relEr + (long)nhop[j * 4 + 2] * D;
#pragma unroll
    for (int i = 0; i < NS; ++i) { int dd = lane + i * 32; m[i] = (dd < D) ? er0[dd] + er1[dd] : 0.f; }
  }
  const float* xsr = xs + (long)s * D;
  const float* xdr = xd + (long)d * D;
  float sp = 0.f;
#pragma unroll
  for (int i = 0; i < NS; ++i) {
    int dd = lane + i * 32;
    if (dd < D) { m[i] += xsr[dd] + xdr[dd]; sp += m[i] * a2[dd]; }
  }
  for (int off = 16; off; off >>= 1) sp += __shfl_xor(sp, off);
  const float lk = sp > 0.f ? sp : ALPHA_LRELU * sp;
  const float w = __expf(-lk);
  if (lane == 0) atomicAdd(&denom[s], w);
  float* hrow = H + ((long)s * nslots + slot) * D;
#pragma unroll
  for (int i = 0; i < NS; ++i) {
    int dd = lane + i * 32;
    if (dd < D) atomicAdd(&hrow[dd], w * m[i]);
  }
}

template <int D>
__global__ __launch_bounds__(256) void gat_edge_new(
    const float* __restrict__ xs, const float* __restrict__ xd,
    const float* __restrict__ relEr,
    const int* __restrict__ srcA, const int* __restrict__ dstA,
    const int* __restrict__ etype, int En,
    const float* __restrict__ otherN, const float* __restrict__ wvec,  // bias = otherN[e,:DI].wvec
    const float* __restrict__ a2, float* __restrict__ H,
    float* __restrict__ denom, int slot, int nslots) {
  const int wave = threadIdx.x >> 5, lane = threadIdx.x & 31;
  const long e = (long)blockIdx.x * (blockDim.x >> 5) + wave;
  if (e >= En) return;
  constexpr int NS = (D + 31) / 32;
  const int s = srcA[e], d = dstA[e];
  const float* er = relEr + (long)etype[e] * D;
  const float* xsr = xs + (long)s * D;
  const float* xdr = xd + (long)d * D;
  float m[NS];
  float sp = 0.f;
#pragma unroll
  for (int i = 0; i < NS; ++i) {
    int dd = lane + i * 32;
    float v = 0.f;
    if (dd < D) { v = er[dd] + xsr[dd] + xdr[dd]; sp += v * a2[dd]; }
    m[i] = v;
  }
  // fold per-edge scalar bias = dot(otherN[e], wvec) into the same reduction
  const float* orow = otherN + (long)e * DI;
  for (int dd = lane; dd < DI; dd += 32) sp += orow[dd] * wvec[dd];
  for (int off = 16; off; off >>= 1) sp += __shfl_xor(sp, off);
  const float lk = sp > 0.f ? sp : ALPHA_LRELU * sp;
  const float w = __expf(-lk);
  if (lane == 0) atomicAdd(&denom[s], w);
  float* hrow = H + ((long)s * nslots + slot) * D;
#pragma unroll
  for (int i = 0; i < NS; ++i) {
    int dd = lane + i * 32;
    if (dd < D) atomicAdd(&hrow[dd], w * m[i]);
  }
}

// H[n,slot,d] = elu(H[n,slot,d] / (denom[slot,n] + 1e-12))
__global__ void norm_elu(float* __restrict__ H, const float* __restrict__ denom,
                         int N, int nslots, int D) {
  const long idx = (long)blockIdx.x * blockDim.x + threadIdx.x;
  const long total = (long)N * nslots * D;
  if (idx >= total) return;
  const int slot = (int)((idx / D) % nslots);
  const int n = (int)(idx / ((long)nslots * D));
  float v = H[idx] / (denom[(long)slot * N + n] + 1e-12f);
  H[idx] = v > 0.f ? v : (__expf(v) - 1.f);
}

// ---------------------------------------------------------------------------
// Small-sequence multi-head self attention (S<=4): thread per (n, head, s).
// o layout: [n][s*D + h*hd + j]  (== transpose(0,2,1,3).reshape(N,-1))
// ---------------------------------------------------------------------------
__global__ void attn_small(const float* __restrict__ q, const float* __restrict__ k,
                           const float* __restrict__ v, float* __restrict__ o,
                           int N, int S, int nh, int hd, float scale) {
  const long idx = (long)blockIdx.x * blockDim.x + threadIdx.x;
  const long total = (long)N * nh * S;
  if (idx >= total) return;
  const int s = (int)(idx % S);
  const int h = (int)((idx / S) % nh);
  const long n = idx / ((long)S * nh);
  const int D = nh * hd;
  const float* qr = q + (n * S + s) * (long)D + h * hd;
  float sc[4];
  float mx = -1e30f;
  for (int t = 0; t < S; ++t) {
    const float* kr = k + (n * S + t) * (long)D + h * hd;
    float acc = 0.f;
    for (int j = 0; j < hd; ++j) acc += qr[j] * kr[j];
    sc[t] = acc * scale;
    mx = fmaxf(mx, sc[t]);
  }
  float sum = 0.f;
  for (int t = 0; t < S; ++t) { sc[t] = __expf(sc[t] - mx); sum += sc[t]; }
  const float inv = 1.f / sum;
  float* orow = o + n * (long)(S * D) + s * D + h * hd;
  for (int j = 0; j < hd; ++j) {
    float acc = 0.f;
    for (int t = 0; t < S; ++t) acc += sc[t] * v[(n * S + t) * (long)D + h * hd + j];
    orow[j] = acc * inv;
  }
}

__global__ void set_mask(const int* __restrict__ batch, float* __restrict__ mask, int B) {
  const int i = blockIdx.x * blockDim.x + threadIdx.x;
  if (i < B) mask[batch[i * 3 + 2]] = 1.0f;
}

// out[n,:] = l2norm(entW[n,:] + mask[n]*x2[n,:]) , D=200, wave per row
__global__ void final_out_kernel(const float* __restrict__ entW,
                                 const float* __restrict__ x2,
                                 const float* __restrict__ mask,
                                 float* __restrict__ out, int N) {
  const int D = DO;
  const int wave = threadIdx.x >> 5, lane = threadIdx.x & 31;
  const long row = (long)blockIdx.x * (blockDim.x >> 5) + wave;
  if (row >= N) return;
  const float mk = mask[row];
  float vals[7];
  float ss = 0.f;
#pragma unroll
  for (int i = 0; i < 7; ++i) {
    const int d = lane + i * 32;
    float v = 0.f;
    if (d < D) v = entW[row * D + d] + mk * x2[row * D + d];
    vals[i] = v;
    ss += v * v;
  }
  for (int off = 16; off; off >>= 1) ss += __shfl_xor(ss, off);
  const float inv = 1.f / fmaxf(sqrtf(ss), 1e-12f);
#pragma unroll
  for (int i = 0; i < 7; ++i) {
    const int d = lane + i * 32;
    if (d < D) out[row * D + d] = vals[i] * inv;
  }
}

// ---------------------------------------------------------------------------
// Workspace layout (floats). Regions for layer-1 and layer-2 are reused.
// Total ~158M floats (~632 MB).
// ---------------------------------------------------------------------------
#define OFF_ENT    0UL                        // 2.5M  : l2norm(entity_emb)
#define OFF_OTHER  2500000UL                  // 5M    : l2norm(new_edge_other)
#define OFF_XS     7500000UL                  // 10M   : xs heads (L1: 2x5M) / xs2 (L2)
#define OFF_XD     17500000UL                 // 10M   : xd heads / xd2
#define OFF_RELER1 27500000UL                 // 0.1M  : 2 x [500,100]
#define OFF_RELER2 27600000UL                 // 0.1M  : [500,200]
#define OFF_H      27700000UL                 // 20M   : H1 [50000,4,100] / H2 [50000,2,200]
#define OFF_DEN    47700000UL                 // 0.2M  : denom slots
#define OFF_Q      47900000UL                 // 20M
#define OFF_K      67900000UL                 // 20M
#define OFF_V      87900000UL                 // 20M
#define OFF_O      107900000UL                // 20M
#define OFF_X      127900000UL                // 10M   : x after SA1 [50000,200]
#define OFF_X2     137900000UL                // 10M   : x2 after SA2
#define OFF_ENTW   147900000UL                // 10M   : ent_n @ W_entities
#define OFF_MASK   157900000UL                // 50K

extern "C" void kernel_launch(void* const* d_in, const int* in_sizes, int n_in,
                              void* d_out, int out_size, void* d_ws, size_t ws_size,
                              hipStream_t stream) {
  const float* entity_emb   = (const float*)d_in[0];
  const float* relation_emb = (const float*)d_in[1];
  const float* W_gat        = (const float*)d_in[2];
  const float* W_entities   = (const float*)d_in[3];
  const float* a1           = (const float*)d_in[4];   // [2,100,150]
  const float* a2_1         = (const float*)d_in[5];   // [2,100]
  const float* wother1      = (const float*)d_in[6];   // [2,50]
  const float* a_out        = (const float*)d_in[7];   // [200,600]
  const float* a2_out       = (const float*)d_in[8];   // [200]
  const float* wother_out   = (const float*)d_in[9];   // [50]
  const float* Wq1          = (const float*)d_in[10];
  const float* Wk1          = (const float*)d_in[11];
  const float* Wv1          = (const float*)d_in[12];
  const float* Wout1        = (const float*)d_in[13];  // [400,200]
  const float* Wq2          = (const float*)d_in[14];
  const float* Wk2          = (const float*)d_in[15];
  const float* Wv2          = (const float*)d_in[16];
  const float* Wout2        = (const float*)d_in[17];  // [400,200]
  const int*   batch_inputs = (const int*)d_in[18];    // [4096,3]
  const int*   edge_list    = (const int*)d_in[19];    // [2,E]
  const int*   edge_type    = (const int*)d_in[20];
  const int*   new_edge_list= (const int*)d_in[21];    // [2,E_NEW]
  const int*   new_edge_type= (const int*)d_in[22];
  const float* new_edge_oth = (const float*)d_in[23];  // [E_NEW,50]
  const int*   nhop         = (const int*)d_in[24];    // [E_NHOP,4]

  float* ws      = (float*)d_ws;
  float* ent_n   = ws + OFF_ENT;
  float* other_n = ws + OFF_OTHER;
  float* xsB     = ws + OFF_XS;
  float* xdB     = ws + OFF_XD;
  float* relEr1  = ws + OFF_RELER1;
  float* relEr2  = ws + OFF_RELER2;
  float* H       = ws + OFF_H;
  float* den     = ws + OFF_DEN;
  float* qB      = ws + OFF_Q;
  float* kB      = ws + OFF_K;
  float* vB      = ws + OFF_V;
  float* oB      = ws + OFF_O;
  float* xB      = ws + OFF_X;
  float* x2B     = ws + OFF_X2;
  float* entW    = ws + OFF_ENTW;
  float* mask    = ws + OFF_MASK;

  float* out_ent = (float*)d_out;                       // [50000,200]
  float* out_rel = (float*)d_out + (size_t)NN * DO;     // [500,200]

  const int* srcM = edge_list;                // edge_list[0]
  const int* dstM = edge_list + NE_MAIN;      // edge_list[1]
  const int* srcN = new_edge_list;
  const int* dstN = new_edge_list + NE_NEW;

  // ---- Phase A: row l2 norms ----
  l2norm_rows<<<ceil_div(NN, 8), 256, 0, stream>>>(entity_emb, ent_n, NN, DI);
  l2norm_rows<<<ceil_div(NE_NEW, 8), 256, 0, stream>>>(new_edge_oth, other_n, NE_NEW, DI);

  // ---- Phase B: layer-1 projections (WMMA, B^T slices of a1 with ldb=150) ----
  for (int h = 0; h < 2; ++h) {
    const float* a1h = a1 + (size_t)h * DH * (3 * DI);
    launch_gemm(true, ent_n, a1h,        xsB + (size_t)h * NN * DH, NN, DH, DI, DI, 3 * DI, DH, stream);
    launch_gemm(true, ent_n, a1h + DI,   xdB + (size_t)h * NN * DH, NN, DH, DI, DI, 3 * DI, DH, stream);
    launch_gemm(true, relation_emb, a1h + 2 * DI, relEr1 + (size_t)h * NR * DH, NR, DH, DI, DI, 3 * DI, DH, stream);
  }

  // ---- Phase C: layer-1 edge scatter (4 slots: head0, head1, nhead0, nhead1) ----
  hipMemsetAsync(H, 0, (size_t)NN * 4 * DH * sizeof(float), stream);
  hipMemsetAsync(den, 0, (size_t)4 * NN * sizeof(float), stream);
  hipMemsetAsync(mask, 0, (size_t)NN * sizeof(float), stream);
  const int ETOT = NE_MAIN + NE_NHOP;
  for (int h = 0; h < 2; ++h) {
    gat_edge_main<DH><<<ceil_div(ETOT, 8), 256, 0, stream>>>(
        xsB + (size_t)h * NN * DH, xdB + (size_t)h * NN * DH, relEr1 + (size_t)h * NR * DH,
        srcM, dstM, edge_type, NE_MAIN, nhop, NE_NHOP,
        a2_1 + h * DH, H, den + (size_t)h * NN, h, 4);
    gat_edge_new<DH><<<ceil_div(NE_NEW, 8), 256, 0, stream>>>(
        xsB + (size_t)h * NN * DH, xdB + (size_t)h * NN * DH, relEr1 + (size_t)h * NR * DH,
        srcN, dstN, new_edge_type, NE_NEW, other_n, wother1 + h * DI,
        a2_1 + h * DH, H, den + (size_t)(2 + h) * NN, 2 + h, 4);
  }
  norm_elu<<<ceil_div(NN * 4 * DH, 256), 256, 0, stream>>>(H, den, NN, 4, DH);

  // ---- Phase D: self-attention 1 (S=4, nh=4, hd=25); X1 = H viewed [200000,100] ----
  launch_gemm(false, H, Wq1, qB, NN * 4, DH, DH, DH, DH, DH, stream);
  launch_gemm(false, H, Wk1, kB, NN * 4, DH, DH, DH, DH, DH, stream);
  launch_gemm(false, H, Wv1, vB, NN * 4, DH, DH, DH, DH, DH, stream);
  attn_small<<<ceil_div(NN * 4 * 4, 256), 256, 0, stream>>>(qB, kB, vB, oB, NN, 4, 4, 25, 0.2f);
  launch_gemm(false, oB, Wout1, xB, NN, DO, 400, 400, DO, DO, stream);  // x [50000,200]

  // ---- Phase E: out_rel, layer-2 projections ----
  launch_gemm(false, relation_emb, W_gat, out_rel, NR, DO, DI, DI, DO, DO, stream);
  launch_gemm(true, out_rel, a_out + 2 * DO, relEr2, NR, DO, DO, DO, 3 * DO, DO, stream);
  launch_gemm(true, xB, a_out,       xsB, NN, DO, DO, DO, 3 * DO, DO, stream);
  launch_gemm(true, xB, a_out + DO,  xdB, NN, DO, DO, DO, 3 * DO, DO, stream);

  // ---- Phase F: layer-2 edge scatter (2 slots: main, new) ----
  hipMemsetAsync(H, 0, (size_t)NN * 2 * DO * sizeof(float), stream);
  hipMemsetAsync(den, 0, (size_t)2 * NN * sizeof(float), stream);
  gat_edge_main<DO><<<ceil_div(ETOT, 8), 256, 0, stream>>>(
      xsB, xdB, relEr2, srcM, dstM, edge_type, NE_MAIN, nhop, NE_NHOP,
      a2_out, H, den, 0, 2);
  gat_edge_new<DO><<<ceil_div(NE_NEW, 8), 256, 0, stream>>>(
      xsB, xdB, relEr2, srcN, dstN, new_edge_type, NE_NEW, other_n, wother_out,
      a2_out, H, den + (size_t)NN, 1, 2);
  norm_elu<<<ceil_div(NN * 2 * DO, 256), 256, 0, stream>>>(H, den, NN, 2, DO);

  // ---- Phase G: self-attention 2 (S=2, nh=4, hd=50); X2 = H viewed [100000,200] ----
  launch_gemm(false, H, Wq2, qB, NN * 2, DO, DO, DO, DO, DO, stream);
  launch_gemm(false, H, Wk2, kB, NN * 2, DO, DO, DO, DO, DO, stream);
  launch_gemm(false, H, Wv2, vB, NN * 2, DO, DO, DO, DO, DO, stream);
  attn_small<<<ceil_div(NN * 4 * 2, 256), 256, 0, stream>>>(qB, kB, vB, oB, NN, 2, 4, 50,
                                                            0.14142135623730951f);  // 1/sqrt(50)
  launch_gemm(false, oB, Wout2, x2B, NN, DO, 400, 400, DO, DO, stream);

  // ---- Phase H: residual + mask + final l2norm ----
  launch_gemm(false, ent_n, W_entities, entW, NN, DO, DI, DI, DO, DO, stream);
  set_mask<<<ceil_div(NBATCH, 256), 256, 0, stream>>>(batch_inputs, mask, NBATCH);
  final_out_kernel<<<ceil_div(NN, 8), 256, 0, stream>>>(entW, x2B, mask, out_ent, NN);
}